// Parking_Trajectory_Planner_31671088841399
// MI455X (gfx1250) — compile-verified
//
#include <hip/hip_runtime.h>
#include <math.h>

// ---------------------------------------------------------------------------
// Parking trajectory planner for MI455X (gfx1250, wave32).
// - Heavy GEMMs on V_WMMA_F32_16X16X4_F32 (full f32 precision).
// - Attention collapsed to rank-2 via fused weights (G 2x2, Dv 256x2).
// - Map crop/sort: histogram-select + bitonic sort, one WG per batch.
// - W_map1^T streamed into LDS with double-buffered async LDS-DMA
//   (global_load_async_to_lds_b128 / s_wait_asynccnt).
// ---------------------------------------------------------------------------

typedef __attribute__((ext_vector_type(2))) float v2f;
typedef __attribute__((ext_vector_type(8))) float v8f;

#define NPTS   20000
#define NBINS  4096
#define CAP    1024
#define NS_    4
#define A_     5
#define OFF_VAR   7680
#define OFF_VIEWS 15360

__device__ __forceinline__ v8f wmma4(v2f a, v2f b, v8f c) {
  // D = A(16x4, f32) x B(4x16, f32) + C(16x16, f32)
  return __builtin_amdgcn_wmma_f32_16x16x4_f32(false, a, false, b, (short)0, c,
                                               false, false);
}
__device__ __forceinline__ float sigf(float x) { return 1.0f / (1.0f + expf(-x)); }

// Async copy of 16 bytes global -> LDS (CDNA5 LDS-DMA, tracked by ASYNCcnt).
__device__ __forceinline__ void async_g2l_b128(unsigned lds_byte_addr,
                                               const void* gaddr) {
  asm volatile("global_load_async_to_lds_b128 %0, %1, off"
               :
               : "v"(lds_byte_addr), "v"(gaddr)
               : "memory");
}
__device__ __forceinline__ void wait_async_le4() {
  asm volatile("s_wait_asynccnt 0x4" ::: "memory");
}
__device__ __forceinline__ void wait_async_0() {
  asm volatile("s_wait_asynccnt 0x0" ::: "memory");
}

__device__ float blockSum(float v, float* red) {
  int t = threadIdx.x;
  red[t] = v; __syncthreads();
  for (int off = 128; off > 0; off >>= 1) {
    if (t < off) red[t] += red[t + off];
    __syncthreads();
  }
  float r = red[0]; __syncthreads();
  return r;
}

// --------------------------------------------------------------------------
// Fuse per-step weights: Cq=wq@Wq etc (256x2), G2 = scale*Cq^T Ck (2x2),
// Dv = W_out@Cv (256x2), and WT = W_map1^T for coalesced WMMA B loads.
// --------------------------------------------------------------------------
__global__ __launch_bounds__(256) void fuse_kernel(
    const float* __restrict__ Win, const float* __restrict__ Wq,
    const float* __restrict__ Wk, const float* __restrict__ Wv,
    const float* __restrict__ Wout, const float* __restrict__ Wmap1,
    float* g2, float* dv, float* wt) {
  int s = blockIdx.x, t = threadIdx.x;
  __shared__ float lCq[512], lCk[512], lCv[512];
  const float* Wi = Win + (size_t)s * 768 * 256;
  const float* q = Wq + s * 512;
  const float* k = Wk + s * 512;
  const float* v = Wv + s * 512;
  float cq0 = 0, cq1 = 0, ck0 = 0, ck1 = 0, cv0 = 0, cv1 = 0;
  for (int kk = 0; kk < 256; ++kk) {
    float a = Wi[t * 256 + kk];
    float b = Wi[(256 + t) * 256 + kk];
    float c = Wi[(512 + t) * 256 + kk];
    cq0 += a * q[kk * 2]; cq1 += a * q[kk * 2 + 1];
    ck0 += b * k[kk * 2]; ck1 += b * k[kk * 2 + 1];
    cv0 += c * v[kk * 2]; cv1 += c * v[kk * 2 + 1];
  }
  lCq[t * 2] = cq0; lCq[t * 2 + 1] = cq1;
  lCk[t * 2] = ck0; lCk[t * 2 + 1] = ck1;
  lCv[t * 2] = cv0; lCv[t * 2 + 1] = cv1;
  __syncthreads();
  const float* Wo = Wout + (size_t)s * 65536;
  float d0 = 0, d1 = 0;
  for (int kk = 0; kk < 256; ++kk) {
    float w = Wo[t * 256 + kk];
    d0 += w * lCv[kk * 2]; d1 += w * lCv[kk * 2 + 1];
  }
  dv[s * 512 + t * 2] = d0; dv[s * 512 + t * 2 + 1] = d1;
  if (t < 4) {
    int a = t >> 1, b = t & 1;
    float g = 0;
    for (int kk = 0; kk < 256; ++kk) g += lCq[kk * 2 + a] * lCk[kk * 2 + b];
    g2[s * 4 + t] = g * 0.0625f;  // scale = 1/sqrt(256)
  }
  const float* Wm = Wmap1 + (size_t)s * 65536;
  float* wts = wt + (size_t)s * 65536;
  for (int idx = t; idx < 65536; idx += 256) {
    int tt = idx >> 8, qq = idx & 255;
    wts[qq * 256 + tt] = Wm[tt * 256 + qq];
  }
}

__global__ void init_anchor_kernel(const float* __restrict__ inp, float* anchor) {
  int i = blockIdx.x * blockDim.x + threadIdx.x;
  if (i < 384) anchor[i] = inp[i];
}

// Reset LSTM state to 1.0 each step; emit pre_mean/pre_var slot 0.
__global__ void step_init_kernel(float* h0, float* h1, float* c0, float* c1,
                                 const float* __restrict__ anchor, float* out, int step) {
  int gid = blockIdx.x * blockDim.x + threadIdx.x;
  int stride = gridDim.x * blockDim.x;
  for (int i = gid; i < 32768; i += stride) {
    h0[i] = 1.0f; h1[i] = 1.0f; c0[i] = 1.0f; c1[i] = 1.0f;
  }
  if (gid < 384) {
    int b = gid / 3, l = gid % 3;
    size_t idx = ((size_t)(b * NS_ + step) * A_ + 0) * 3 + l;
    out[idx] = anchor[gid];
    out[OFF_VAR + idx] = 0.01f;
  }
}

// --------------------------------------------------------------------------
// _cal_map_last: rotate map points, keep the 256 smallest per axis (sorted),
// pad with 0. Histogram select + gather + bitonic sort, one WG per batch.
// --------------------------------------------------------------------------
__global__ __launch_bounds__(256) void map_select_kernel(
    const float* __restrict__ mapPts, const float* __restrict__ anchor,
    float* mlOut, float* out, int step, int aidx) {
  int b = blockIdx.x, tid = threadIdx.x;
  __shared__ unsigned hist[2][NBINS];
  __shared__ float buf[2][CAP];
  __shared__ unsigned scan[256];
  __shared__ unsigned targ[2];
  __shared__ int thrBin[2];
  __shared__ unsigned gcnt[2];

  for (int i = tid; i < 2 * NBINS; i += 256) ((unsigned*)hist)[i] = 0u;
  float px0 = anchor[b * 3], py0 = anchor[b * 3 + 1], th = anchor[b * 3 + 2];
  float ct = cosf(th), st = sinf(th);
  __syncthreads();

  for (int i = tid; i < NPTS; i += 256) {
    float dx = mapPts[i] - px0, dy = mapPts[NPTS + i] - py0;
    float m0 = ct * dx + st * dy, m1 = -st * dx + ct * dy;
    float d = sqrtf(m0 * m0 + m1 * m1);
    if (d < 10.0f) {
      int b0 = (int)((m0 + 10.0f) * 204.8f); b0 = b0 < 0 ? 0 : (b0 > 4095 ? 4095 : b0);
      int b1 = (int)((m1 + 10.0f) * 204.8f); b1 = b1 < 0 ? 0 : (b1 > 4095 ? 4095 : b1);
      atomicAdd(&hist[0][b0], 1u);
      atomicAdd(&hist[1][b1], 1u);
    }
  }
  __syncthreads();

  for (int ax = 0; ax < 2; ++ax) {
    unsigned seg = 0;
    for (int k2 = 0; k2 < 16; ++k2) seg += hist[ax][tid * 16 + k2];
    scan[tid] = seg; __syncthreads();
    for (int off = 1; off < 256; off <<= 1) {
      unsigned v = scan[tid];
      unsigned ad = (tid >= off) ? scan[tid - off] : 0u;
      __syncthreads();
      scan[tid] = v + ad; __syncthreads();
    }
    unsigned total = scan[255];
    unsigned target = total < 256u ? total : 256u;
    if (tid == 0) { targ[ax] = target; thrBin[ax] = -1; }
    __syncthreads();
    if (target > 0u) {
      unsigned lo = tid ? scan[tid - 1] : 0u;
      if (lo < target && scan[tid] >= target) {
        unsigned cum = lo;
        for (int k2 = 0; k2 < 16; ++k2) {
          cum += hist[ax][tid * 16 + k2];
          if (cum >= target) { thrBin[ax] = tid * 16 + k2; break; }
        }
      }
    }
    __syncthreads();
  }

  if (tid == 0) { gcnt[0] = 0u; gcnt[1] = 0u; }
  __syncthreads();
  int T0 = thrBin[0], T1 = thrBin[1];
  for (int i = tid; i < NPTS; i += 256) {
    float dx = mapPts[i] - px0, dy = mapPts[NPTS + i] - py0;
    float m0 = ct * dx + st * dy, m1 = -st * dx + ct * dy;
    float d = sqrtf(m0 * m0 + m1 * m1);
    if (d < 10.0f) {
      int b0 = (int)((m0 + 10.0f) * 204.8f); b0 = b0 < 0 ? 0 : (b0 > 4095 ? 4095 : b0);
      int b1 = (int)((m1 + 10.0f) * 204.8f); b1 = b1 < 0 ? 0 : (b1 > 4095 ? 4095 : b1);
      if (b0 <= T0) { unsigned p = atomicAdd(&gcnt[0], 1u); if (p < CAP) buf[0][p] = m0; }
      if (b1 <= T1) { unsigned p = atomicAdd(&gcnt[1], 1u); if (p < CAP) buf[1][p] = m1; }
    }
  }
  __syncthreads();
  unsigned c0n = gcnt[0] < CAP ? gcnt[0] : CAP;
  unsigned c1n = gcnt[1] < CAP ? gcnt[1] : CAP;
  for (int i = tid; i < CAP; i += 256) {
    if (i >= (int)c0n) buf[0][i] = 1e10f;
    if (i >= (int)c1n) buf[1][i] = 1e10f;
  }
  __syncthreads();

  for (int kk = 2; kk <= CAP; kk <<= 1)
    for (int j = kk >> 1; j > 0; j >>= 1) {
      for (int i = tid; i < CAP; i += 256) {
        int ixj = i ^ j;
        if (ixj > i) {
          bool asc = ((i & kk) == 0);
          float x0 = buf[0][i], x1 = buf[0][ixj];
          if ((x0 > x1) == asc) { buf[0][i] = x1; buf[0][ixj] = x0; }
          float y0 = buf[1][i], y1 = buf[1][ixj];
          if ((y0 > y1) == asc) { buf[1][i] = y1; buf[1][ixj] = y0; }
        }
      }
      __syncthreads();
    }

  int m = tid;
  float v0 = (m < (int)targ[0]) ? buf[0][m] : 0.0f;
  float v1 = (m < (int)targ[1]) ? buf[1][m] : 0.0f;
  size_t vb = OFF_VIEWS + ((size_t)(b * NS_ + step) * A_ + aidx) * 512;
  out[vb + m] = v0;
  out[vb + 256 + m] = v1;
  if (mlOut) {
    mlOut[b * 512 + m * 2] = v0;
    mlOut[b * 512 + m * 2 + 1] = v1;
  }
}

// --------------------------------------------------------------------------
// Rank-2 attention + map-MLP (WMMA f32, async double-buffered B panels)
// + both encoder LayerNorms -> xl.
// --------------------------------------------------------------------------
__global__ __launch_bounds__(256) void attn_kernel(
    const float* __restrict__ ml, const float* __restrict__ G2s,
    const float* __restrict__ Dvs, const float* __restrict__ WT,
    const float* __restrict__ Wm2s, const float* __restrict__ Wencs,
    const float* __restrict__ anchor, const float* __restrict__ endp,
    float* xl) {
  int b = blockIdx.x, tid = threadIdx.x;
  __shared__ float mlx[256], mly[256], lA0[256], lA1[256];
  __shared__ float lD0[256], lD1[256], x2[256], enc[256], red[256], lW2[256];
  // Double-buffered 256x16 column panels of WT (16 KB each).
  __shared__ __align__(16) float panel[2][4096];

  mlx[tid] = ml[b * 512 + tid * 2];
  mly[tid] = ml[b * 512 + tid * 2 + 1];
  lD0[tid] = Dvs[tid * 2];
  lD1[tid] = Dvs[tid * 2 + 1];
  lW2[tid] = Wm2s[tid];

  // Kick off async DMA of panel 0 (overlaps with softmax phase below).
  {
#pragma unroll
    for (int i = 0; i < 4; ++i) {
      int cid = tid + 256 * i;           // 1024 x b128 chunks per panel
      int row = cid >> 2;                // q row 0..255
      int c4 = (cid & 3) * 4;            // float col within 16-wide panel
      const float* g = WT + (size_t)row * 256 + 0 * 16 + c4;
      unsigned lds = (unsigned)(uintptr_t)&panel[0][row * 16 + c4];
      async_g2l_b128(lds, (const void*)g);
    }
  }

  float a0 = anchor[b * 3], a1 = anchor[b * 3 + 1], a2 = anchor[b * 3 + 2];
  float in9[9] = {a0, a1, a2, a0 - endp[0], a1 - endp[1], a2 - endp[2],
                  0.01f, 0.01f, 0.01f};
  float er = 0;
#pragma unroll
  for (int k2 = 0; k2 < 9; ++k2) er += in9[k2] * Wencs[tid * 9 + k2];
  enc[tid] = er;
  __syncthreads();

  // softmax(ml G ml^T) @ ml  (rank-2 logits, fully in LDS)
  float g00 = G2s[0], g01 = G2s[1], g10 = G2s[2], g11 = G2s[3];
  float xi = mlx[tid], yi = mly[tid];
  float u0 = xi * g00 + yi * g10, u1 = xi * g01 + yi * g11;
  float mx = -1e30f;
  for (int j = 0; j < 256; ++j) mx = fmaxf(mx, u0 * mlx[j] + u1 * mly[j]);
  float sum = 0, aa0 = 0, aa1 = 0;
  for (int j = 0; j < 256; ++j) {
    float e = expf(u0 * mlx[j] + u1 * mly[j] - mx);
    sum += e; aa0 += e * mlx[j]; aa1 += e * mly[j];
  }
  float inv = 1.0f / sum;
  lA0[tid] = aa0 * inv; lA1[tid] = aa1 * inv;
  __syncthreads();

  // x1 = relu(X) @ W_map1^T with X[s,q]=relu(A0[q]D0[s]+A1[q]D1[s]);
  // x2[s] = sum_t relu(x1[s,t]) * w2[t]
  int wave = tid >> 5, lane = tid & 31, n_l = lane & 15, kh = lane >> 4;
  int s0 = wave * 32, s1 = s0 + 16;
  float dA0 = lD0[s0 + n_l], dB0 = lD1[s0 + n_l];
  float dA1 = lD0[s1 + n_l], dB1 = lD1[s1 + n_l];
  float part0[8], part1[8];
#pragma unroll
  for (int r = 0; r < 8; ++r) { part0[r] = 0; part1[r] = 0; }

  for (int tt = 0; tt < 16; ++tt) {
    int cur = tt & 1;
    if (tt < 15) {
      // Prefetch next panel into the other buffer (its last readers finished
      // at the end-of-iteration barrier of tt-1).
#pragma unroll
      for (int i = 0; i < 4; ++i) {
        int cid = tid + 256 * i;
        int row = cid >> 2;
        int c4 = (cid & 3) * 4;
        const float* g = WT + (size_t)row * 256 + (tt + 1) * 16 + c4;
        unsigned lds = (unsigned)(uintptr_t)&panel[cur ^ 1][row * 16 + c4];
        async_g2l_b128(lds, (const void*)g);
      }
      wait_async_le4();   // panel tt (issued earlier, in-order) is resident
    } else {
      wait_async_0();
    }
    __syncthreads();

    v8f acc0{}, acc1{};
    const float* P = &panel[cur][0];
    for (int kb = 0; kb < 64; ++kb) {
      int q0 = kb * 4 + kh * 2;
      float A0a = lA0[q0], A1a = lA1[q0], A0b = lA0[q0 + 1], A1b = lA1[q0 + 1];
      v2f bv; bv.x = P[q0 * 16 + n_l]; bv.y = P[(q0 + 1) * 16 + n_l];
      v2f av0; av0.x = fmaxf(dA0 * A0a + dB0 * A1a, 0.0f);
               av0.y = fmaxf(dA0 * A0b + dB0 * A1b, 0.0f);
      v2f av1; av1.x = fmaxf(dA1 * A0a + dB1 * A1a, 0.0f);
               av1.y = fmaxf(dA1 * A0b + dB1 * A1b, 0.0f);
      acc0 = wmma4(av0, bv, acc0);
      acc1 = wmma4(av1, bv, acc1);
    }
    float w2v = lW2[tt * 16 + n_l];
#pragma unroll
    for (int r = 0; r < 8; ++r) {
      part0[r] += fmaxf(acc0[r], 0.0f) * w2v;
      part1[r] += fmaxf(acc1[r], 0.0f) * w2v;
    }
    __syncthreads();  // all waves done reading panel[cur] before overwrite
  }

#pragma unroll
  for (int r = 0; r < 8; ++r) {  // deterministic cross-lane column reduction
    float v0 = part0[r], v1 = part1[r];
    for (int off = 1; off < 16; off <<= 1) {
      v0 += __shfl_xor(v0, off, 32);
      v1 += __shfl_xor(v1, off, 32);
    }
    if (n_l == 0) {
      int rb = (lane >= 16) ? 8 : 0;
      x2[s0 + rb + r] = v0;
      x2[s1 + rb + r] = v1;
    }
  }
  __syncthreads();

  float xv = x2[tid];
  float mu = blockSum(xv, red) * (1.0f / 256.0f);
  float dm = xv - mu;
  float var = blockSum(dm * dm, red) * (1.0f / 256.0f);
  float em = dm * rsqrtf(var + 1e-5f);
  float ev = enc[tid];
  float mu2 = blockSum(ev, red) * (1.0f / 256.0f);
  float d2 = ev - mu2;
  float var2 = blockSum(d2 * d2, red) * (1.0f / 256.0f);
  float ea = d2 * rsqrtf(var2 + 1e-5f);
  xl[b * 512 + tid] = ea;
  xl[b * 512 + 256 + tid] = em;
}

// --------------------------------------------------------------------------
// One LSTM layer (batched WMMA GEMM: [128,Kx]@Wih^T + [128,256]@Whh^T, gates).
// Grid: 16 j-tile blocks; 8 waves = 8 batch m-tiles.
// --------------------------------------------------------------------------
__global__ __launch_bounds__(256) void lstm_kernel(
    const float* __restrict__ X, int Kx, const float* __restrict__ Wih,
    const float* __restrict__ Hin, const float* __restrict__ Whh,
    float* C, float* Hout) {
  int jt = blockIdx.x, tid = threadIdx.x;
  int wave = tid >> 5, lane = tid & 31, n_l = lane & 15, kh = lane >> 4;
  int mrow = wave * 16 + n_l;
  int ncolBase = jt * 16 + n_l;
  v8f acc[4] = {};
  const float* xrow = X + (size_t)mrow * Kx;
  for (int kb = 0; kb < (Kx >> 2); ++kb) {
    int k0 = kb * 4 + kh * 2;
    v2f a; a.x = xrow[k0]; a.y = xrow[k0 + 1];
#pragma unroll
    for (int g = 0; g < 4; ++g) {
      const float* wr = Wih + (size_t)(g * 256 + ncolBase) * Kx;
      v2f bv; bv.x = wr[k0]; bv.y = wr[k0 + 1];
      acc[g] = wmma4(a, bv, acc[g]);
    }
  }
  const float* hrow = Hin + (size_t)mrow * 256;
  for (int kb = 0; kb < 64; ++kb) {
    int k0 = kb * 4 + kh * 2;
    v2f a; a.x = hrow[k0]; a.y = hrow[k0 + 1];
#pragma unroll
    for (int g = 0; g < 4; ++g) {
      const float* wr = Whh + (size_t)(g * 256 + ncolBase) * 256;
      v2f bv; bv.x = wr[k0]; bv.y = wr[k0 + 1];
      acc[g] = wmma4(a, bv, acc[g]);
    }
  }
  int rb = (lane >= 16) ? 8 : 0;
  int j = jt * 16 + n_l;
#pragma unroll
  for (int r = 0; r < 8; ++r) {
    int brow = wave * 16 + rb + r;
    float iv = sigf(acc[0][r]);
    float fv = sigf(acc[1][r]);
    float gv = tanhf(acc[2][r]);
    float ov = sigf(acc[3][r]);
    float cold = C[brow * 256 + j];
    float cl = fv * cold + iv * gv;
    C[brow * 256 + j] = cl;
    Hout[brow * 256 + j] = ov * tanhf(cl);
  }
}

// --------------------------------------------------------------------------
// Head: LN(h1) -> mean/var MLPs -> update anchor, emit pre_mean/pre_var slots.
// --------------------------------------------------------------------------
__global__ __launch_bounds__(256) void head_kernel(
    const float* __restrict__ h1, const float* __restrict__ Wm1s,
    const float* __restrict__ Wm2s, const float* __restrict__ Wv1s,
    const float* __restrict__ Wv2s, const float* __restrict__ dlm,
    float* anchor, float* out, int step, int ai) {
  int b = blockIdx.x, tid = threadIdx.x;
  __shared__ float rm[256], t1[256], tv1[256], red[256];
  float hv = h1[b * 256 + tid];
  float mu = blockSum(hv, red) * (1.0f / 256.0f);
  float d = hv - mu;
  float var = blockSum(d * d, red) * (1.0f / 256.0f);
  float mn = d * rsqrtf(var + 1e-5f);
  rm[tid] = fmaxf(mn, 0.0f);
  __syncthreads();
  float s1 = 0, s2 = 0;
  for (int k2 = 0; k2 < 256; ++k2) {
    float r = rm[k2];
    s1 += r * Wm1s[tid * 256 + k2];
    s2 += r * Wv1s[tid * 256 + k2];
  }
  t1[tid] = fmaxf(s1, 0.0f);
  tv1[tid] = fmaxf(s2, 0.0f);
  __syncthreads();
  if (tid < 6) {
    int l = tid % 3;
    bool isv = tid >= 3;
    const float* W2 = isv ? Wv2s : Wm2s;
    const float* src = isv ? tv1 : t1;
    float s = 0;
    for (int k2 = 0; k2 < 256; ++k2) s += src[k2] * W2[l * 256 + k2];
    size_t idx = ((size_t)(b * NS_ + step) * A_ + ai) * 3 + l;
    if (!isv) {
      float m = tanhf(s) * dlm[l] + anchor[b * 3 + l];
      out[idx] = m;
      anchor[b * 3 + l] = m;
    } else {
      out[OFF_VAR + idx] = 0.1f * sigf(s);
    }
  }
}

// --------------------------------------------------------------------------
extern "C" void kernel_launch(void* const* d_in, const int* in_sizes, int n_in,
                              void* d_out, int out_size, void* d_ws, size_t ws_size,
                              hipStream_t stream) {
  (void)in_sizes; (void)n_in; (void)out_size; (void)ws_size;
  const float* inp  = (const float*)d_in[0];
  const float* map  = (const float*)d_in[1];
  const float* endp = (const float*)d_in[2];
  const float* dlm  = (const float*)d_in[3];
  const float* Wenc = (const float*)d_in[4];
  const float* Wq   = (const float*)d_in[5];
  const float* Wk   = (const float*)d_in[6];
  const float* Wv   = (const float*)d_in[7];
  const float* Win  = (const float*)d_in[8];
  const float* Wout = (const float*)d_in[9];
  const float* Wm1g = (const float*)d_in[10];   // W_map1
  const float* Wm2g = (const float*)d_in[11];   // W_map2
  const float* Wih0 = (const float*)d_in[12];
  const float* Wih1 = (const float*)d_in[13];
  const float* Whh0 = (const float*)d_in[14];
  const float* Whh1 = (const float*)d_in[15];
  const float* Wm1  = (const float*)d_in[16];
  const float* Wm2  = (const float*)d_in[17];
  const float* Wv1  = (const float*)d_in[18];
  const float* Wv2  = (const float*)d_in[19];
  float* out = (float*)d_out;
  float* ws  = (float*)d_ws;

  float* g2   = ws + 0;        // 16
  float* dv   = ws + 256;      // 2048
  float* wt   = ws + 4096;     // 4*65536
  float* mlb  = ws + 266240;   // 65536
  float* xlb  = ws + 331776;   // 65536
  float* h0b[2] = { ws + 397312, ws + 397312 + 32768 };
  float* h1b[2] = { ws + 462848, ws + 462848 + 32768 };
  float* c0b  = ws + 528384;
  float* c1b  = ws + 561152;
  float* anch = ws + 593920;   // 384  (total ~2.3 MB)

  fuse_kernel<<<4, 256, 0, stream>>>(Win, Wq, Wk, Wv, Wout, Wm1g, g2, dv, wt);
  init_anchor_kernel<<<2, 256, 0, stream>>>(inp, anch);

  for (int s = 0; s < 4; ++s) {
    int hc = 0;
    step_init_kernel<<<64, 256, 0, stream>>>(h0b[0], h1b[0], c0b, c1b, anch, out, s);
    for (int a = 0; a < 5; ++a) {
      map_select_kernel<<<128, 256, 0, stream>>>(
          map, anch, (a < 4) ? mlb : (float*)nullptr, out, s, a);
      if (a < 4) {
        attn_kernel<<<128, 256, 0, stream>>>(
            mlb, g2 + s * 4, dv + s * 512, wt + (size_t)s * 65536,
            Wm2g + s * 256, Wenc + s * 2304, anch, endp, xlb);
        lstm_kernel<<<16, 256, 0, stream>>>(
            xlb, 512, Wih0 + (size_t)s * 524288, h0b[hc],
            Whh0 + (size_t)s * 262144, c0b, h0b[1 - hc]);
        lstm_kernel<<<16, 256, 0, stream>>>(
            h0b[1 - hc], 256, Wih1 + (size_t)s * 262144, h1b[hc],
            Whh1 + (size_t)s * 262144, c1b, h1b[1 - hc]);
        hc = 1 - hc;
        head_kernel<<<128, 256, 0, stream>>>(
            h1b[hc], Wm1 + (size_t)s * 65536, Wm2 + s * 768,
            Wv1 + (size_t)s * 65536, Wv2 + s * 768, dlm, anch, out, s, a + 1);
      }
    }
  }
}